// NonLocalResBlock_76089640616780
// MI455X (gfx1250) — compile-verified
//
#include <hip/hip_runtime.h>

typedef __attribute__((ext_vector_type(2))) float v2f;
typedef __attribute__((ext_vector_type(8))) float v8f;

#define B_  4
#define T_  5
#define C_  3
#define H_  128
#define W_  128
#define D_  60      // C*T*R*R
#define DP  64      // padded feature dim
#define NN  4096    // 64*64 spatial after unshuffle
#define KT  128     // keys per softmax chunk (8 x 16 tiles)

// native base-2 exponential: one v_exp_f32
__device__ __forceinline__ float fast_exp2(float x) {
    return __builtin_amdgcn_exp2f(x);
}

// ---------------------------------------------------------------------------
// Kernel 1: pixel-unshuffle x (B,T,C,128,128) -> Q buffer (B, N=4096, 64) f32,
// zero-padded in features 60..63.  d = (t*3+cc)*4 + ri*2 + rj, n = oh*64+ow,
// source pixel = (2*oh+ri, 2*ow+rj).
// ---------------------------------------------------------------------------
__global__ void nlb_prep_q(const float* __restrict__ x, float* __restrict__ Q) {
    int idx = blockIdx.x * blockDim.x + threadIdx.x;
    if (idx >= B_ * NN * DP) return;
    int dp = idx & 63;
    int n  = (idx >> 6) & (NN - 1);
    int b  = idx >> 18;
    float v = 0.0f;
    if (dp < D_) {
        int tc = dp >> 2;             // t*3 + cc  (0..14)
        int rr = dp & 3;
        int ri = rr >> 1, rj = rr & 1;
        int oh = n >> 6, ow = n & 63;
        int hh = oh * 2 + ri, ww = ow * 2 + rj;
        v = x[((size_t)(b * (T_ * C_) + tc) * H_ + hh) * W_ + ww];
    }
    Q[idx] = v;
}

// ---------------------------------------------------------------------------
// Kernel 2: V = g_w @ xf + g_b  -> Gt buffer stored FEATURE-MAJOR (B, 64, N)
// so PV B-fragments (two adjacent keys of one feature) are contiguous b64s.
// Padded feature rows 60..63 are 0.
// ---------------------------------------------------------------------------
__global__ void nlb_prep_g(const float* __restrict__ Q, const float* __restrict__ gw,
                           const float* __restrict__ gb, float* __restrict__ Gt) {
    int idx = blockIdx.x * blockDim.x + threadIdx.x;
    if (idx >= B_ * DP * NN) return;
    int n = idx & (NN - 1);
    int o = (idx >> 12) & 63;
    int b = idx >> 18;
    float acc = 0.0f;
    if (o < D_) {
        const float* qr = Q + (((size_t)b * NN + n) << 6);
        const float* wr = gw + o * D_;
        acc = gb[o];
#pragma unroll 4
        for (int d = 0; d < D_; ++d) acc += wr[d] * qr[d];
    }
    Gt[idx] = acc;   // Gt[b][o][n], contiguous in n -> coalesced stores
}

// ---------------------------------------------------------------------------
// Kernel 3: fused flash attention.  Q==K (B,N,64), V=Gt (B,64,N) -> Y (B,N,64).
// Block = 128 threads = 4 waves; each wave owns one 16-row query tile.
// Chunk loop over 32 chunks of 128 keys:
//   S(16x128)  = 120 x v_wmma_f32_16x16x4_f32 (8 tiles)
//   ONE online-softmax round: 7-op elementwise max/sum tree across the 8
//   tiles in registers, then a single 4-step shfl_xor butterfly per row.
//   Exponentials are base-2 (v_exp_f32) with log2(e) folded into Q fragments.
//   O(16x64) += 128 x v_wmma_f32_16x16x4_f32, V frags are single b64 loads
//   from the feature-major Gt.  P bounced through LDS (C->A layout).
// ---------------------------------------------------------------------------
__global__ __launch_bounds__(128, 1)
void nlb_attn(const float* __restrict__ Q, const float* __restrict__ Gt,
              float* __restrict__ Y) {
    __shared__ float lds[4 * 16 * KT];           // 32KB: one 16x128 P tile per wave

    const int lane = threadIdx.x & 31;
    const int wave = threadIdx.x >> 5;
    const int m    = lane & 15;        // A-layout row / B-layout col / C-layout col
    const int h    = lane >> 4;        // half-wave select
    const int b    = blockIdx.x >> 6;                    // 64 blocks per batch
    const int q0   = ((blockIdx.x & 63) * 4 + wave) * 16;

    const float* Qb  = Q  + (size_t)b * NN * DP;
    const float* Gtb = Gt + (size_t)b * DP * NN;
    float*       Yb  = Y  + (size_t)b * NN * DP;
    float*       pl  = lds + wave * (16 * KT);

    // V feature rows for the 4 output column tiles (feature = ct*16 + m).
    const float* vrow0 = Gtb + (size_t)( 0 + m) * NN;
    const float* vrow1 = Gtb + (size_t)(16 + m) * NN;
    const float* vrow2 = Gtb + (size_t)(32 + m) * NN;
    const float* vrow3 = Gtb + (size_t)(48 + m) * NN;

    // Q tile in A-fragment layout, pre-scaled by log2(e) so softmax runs in
    // base-2 (v_exp_f32 is a native exp2).
    const float LOG2E = 1.4426950408889634f;
    v2f qa[15];
    {
        const float* qrow = Qb + (size_t)(q0 + m) * DP;
#pragma unroll
        for (int kk = 0; kk < 15; ++kk) {
            v2f q = *(const v2f*)(qrow + 4 * kk + 2 * h);
            qa[kk] = q * LOG2E;
        }
    }

    v8f o0{}, o1{}, o2{}, o3{};
    float mrow[8], lrow[8];
#pragma unroll
    for (int j = 0; j < 8; ++j) { mrow[j] = -__builtin_inff(); lrow[j] = 0.0f; }

    for (int ch = 0; ch < NN / KT; ++ch) {
        const int k0 = ch * KT;

        // ---- prefetch next chunk's K rows and V feature rows into WGP$ ---
        {
            const int kn = (ch + 1 < NN / KT) ? (k0 + KT) : k0;
            const char* pk = (const char*)(Qb + (size_t)kn * DP) + lane * 1024;
            __builtin_prefetch(pk,       0, 3);
            __builtin_prefetch(pk + 256, 0, 3);
            __builtin_prefetch(pk + 512, 0, 3);
            __builtin_prefetch(pk + 768, 0, 3);
            const char* pv0 = (const char*)(Gtb + (size_t)lane        * NN + kn);
            const char* pv1 = (const char*)(Gtb + (size_t)(lane + 32) * NN + kn);
            __builtin_prefetch(pv0,       0, 3);
            __builtin_prefetch(pv0 + 256, 0, 3);
            __builtin_prefetch(pv1,       0, 3);
            __builtin_prefetch(pv1 + 256, 0, 3);
        }

        // ---- S = Q @ K^T for eight 16x16 tiles (16x128 logit block) ------
        v8f s[8];
#pragma unroll
        for (int t = 0; t < 8; ++t) {
            v8f acc{};
            const float* krow = Qb + (size_t)(k0 + t * 16 + m) * DP;
#pragma unroll
            for (int kk = 0; kk < 15; ++kk) {
                v2f kb = *(const v2f*)(krow + 4 * kk + 2 * h);
                acc = __builtin_amdgcn_wmma_f32_16x16x4_f32(
                          false, qa[kk], false, kb, (short)0, acc, false, false);
            }
            s[t] = acc;
        }

        // ---- one online-softmax round over 128 columns -------------------
        // row r = 8*h + j; its 128 logits live in s[0..7][j] across 16 lanes.
#pragma unroll
        for (int j = 0; j < 8; ++j) {
            float tm = fmaxf(fmaxf(fmaxf(s[0][j], s[1][j]), fmaxf(s[2][j], s[3][j])),
                             fmaxf(fmaxf(s[4][j], s[5][j]), fmaxf(s[6][j], s[7][j])));
            tm = fmaxf(tm, __shfl_xor(tm, 1, 16));
            tm = fmaxf(tm, __shfl_xor(tm, 2, 16));
            tm = fmaxf(tm, __shfl_xor(tm, 4, 16));
            tm = fmaxf(tm, __shfl_xor(tm, 8, 16));
            float mn = fmaxf(mrow[j], tm);
            float sc = fast_exp2(mrow[j] - mn);
            float p0 = fast_exp2(s[0][j] - mn);
            float p1 = fast_exp2(s[1][j] - mn);
            float p2 = fast_exp2(s[2][j] - mn);
            float p3 = fast_exp2(s[3][j] - mn);
            float p4 = fast_exp2(s[4][j] - mn);
            float p5 = fast_exp2(s[5][j] - mn);
            float p6 = fast_exp2(s[6][j] - mn);
            float p7 = fast_exp2(s[7][j] - mn);
            s[0][j] = p0; s[1][j] = p1; s[2][j] = p2; s[3][j] = p3;
            s[4][j] = p4; s[5][j] = p5; s[6][j] = p6; s[7][j] = p7;
            float su = ((p0 + p1) + (p2 + p3)) + ((p4 + p5) + (p6 + p7));
            su += __shfl_xor(su, 1, 16);
            su += __shfl_xor(su, 2, 16);
            su += __shfl_xor(su, 4, 16);
            su += __shfl_xor(su, 8, 16);
            lrow[j] = lrow[j] * sc + su;
            mrow[j] = mn;
            o0[j] *= sc; o1[j] *= sc; o2[j] *= sc; o3[j] *= sc;
        }

        // ---- transpose P(16x128) from C-layout to A-layout via LDS -------
#pragma unroll
        for (int t = 0; t < 8; ++t)
#pragma unroll
            for (int j = 0; j < 8; ++j)
                pl[(8 * h + j) * KT + t * 16 + m] = s[t][j];
        asm volatile("" ::: "memory");       // keep DS store->load program order
        // (DS ops are in-order within a wave; barrier pins compiler ordering)

        // ---- O += P @ V  (K-dim = 128 keys -> 32 k-chunks of 4) ----------
#pragma unroll
        for (int kk2 = 0; kk2 < 32; ++kk2) {
            v2f pa = *(const v2f*)(pl + m * KT + 4 * kk2 + 2 * h);
            const int kof = k0 + 4 * kk2 + 2 * h;  // two adjacent keys -> one b64
            v2f vb;
            vb = *(const v2f*)(vrow0 + kof);
            o0 = __builtin_amdgcn_wmma_f32_16x16x4_f32(false, pa, false, vb, (short)0, o0, false, false);
            vb = *(const v2f*)(vrow1 + kof);
            o1 = __builtin_amdgcn_wmma_f32_16x16x4_f32(false, pa, false, vb, (short)0, o1, false, false);
            vb = *(const v2f*)(vrow2 + kof);
            o2 = __builtin_amdgcn_wmma_f32_16x16x4_f32(false, pa, false, vb, (short)0, o2, false, false);
            vb = *(const v2f*)(vrow3 + kof);
            o3 = __builtin_amdgcn_wmma_f32_16x16x4_f32(false, pa, false, vb, (short)0, o3, false, false);
        }
        asm volatile("" ::: "memory");       // LDS reads stay before next stores
    }

    // ---- normalize and write Y (row = q0 + 8h + j, col = ct*16 + m) ------
#pragma unroll
    for (int j = 0; j < 8; ++j) {
        float inv = 1.0f / lrow[j];
        float* yr = Yb + (size_t)(q0 + 8 * h + j) * DP;
        yr[ 0 + m] = o0[j] * inv;
        yr[16 + m] = o1[j] * inv;
        yr[32 + m] = o2[j] * inv;
        yr[48 + m] = o3[j] * inv;
    }
}

// ---------------------------------------------------------------------------
// Kernel 4: Z = pixel_shuffle(w_w @ Y + w_b) + x, written as (B,T,C,128,128).
// ---------------------------------------------------------------------------
__global__ void nlb_out(const float* __restrict__ x, const float* __restrict__ Y,
                        const float* __restrict__ ww, const float* __restrict__ wb,
                        float* __restrict__ Z) {
    int idx = blockIdx.x * blockDim.x + threadIdx.x;
    if (idx >= B_ * T_ * C_ * H_ * W_) return;
    int wcol = idx % W_;  int tmp = idx / W_;
    int hrow = tmp % H_;  tmp /= H_;
    int cc   = tmp % C_;  tmp /= C_;
    int t    = tmp % T_;
    int b    = tmp / T_;
    int ri = hrow & 1, rj = wcol & 1;
    int oh = hrow >> 1, ow = wcol >> 1;
    int o  = (t * C_ + cc) * 4 + ri * 2 + rj;
    int n  = oh * 64 + ow;
    const float* yr = Y + (((size_t)b * NN + n) << 6);
    const float* wr = ww + o * D_;
    float acc = wb[o];
#pragma unroll 4
    for (int d = 0; d < D_; ++d) acc += wr[d] * yr[d];
    Z[idx] = acc + x[idx];
}

// ---------------------------------------------------------------------------
extern "C" void kernel_launch(void* const* d_in, const int* in_sizes, int n_in,
                              void* d_out, int out_size, void* d_ws, size_t ws_size,
                              hipStream_t stream) {
    const float* x  = (const float*)d_in[0];
    const float* gw = (const float*)d_in[1];
    const float* gb = (const float*)d_in[2];
    const float* ww = (const float*)d_in[3];
    const float* wb = (const float*)d_in[4];
    float* Z = (float*)d_out;

    const size_t buf = (size_t)B_ * NN * DP;   // 1,048,576 floats = 4 MB
    float* Qbuf = (float*)d_ws;
    float* Gbuf = Qbuf + buf;                  // feature-major (B,64,N)
    float* Ybuf = Gbuf + buf;

    {   // pixel-unshuffle into padded (B,N,64) layout
        int n = B_ * NN * DP;
        nlb_prep_q<<<(n + 255) / 256, 256, 0, stream>>>(x, Qbuf);
    }
    {   // V = g_w @ xf + g_b, stored feature-major
        int n = B_ * DP * NN;
        nlb_prep_g<<<(n + 255) / 256, 256, 0, stream>>>(Qbuf, gw, gb, Gbuf);
    }
    // fused flash attention: 4 batches * 64 blocks, 4 waves (64 queries) each
    nlb_attn<<<B_ * 64, 128, 0, stream>>>(Qbuf, Gbuf, Ybuf);
    {   // w_w @ Y + w_b, pixel-shuffle, residual
        int n = B_ * T_ * C_ * H_ * W_;
        nlb_out<<<(n + 255) / 256, 256, 0, stream>>>(x, Ybuf, ww, wb, Z);
    }
}